// HANModel_1821066133799
// MI455X (gfx1250) — compile-verified
//
#include <hip/hip_runtime.h>
#include <hip/hip_bf16.h>
#include <math.h>

typedef __attribute__((ext_vector_type(2))) float v2f;
typedef __attribute__((ext_vector_type(8))) float v8f;

#define NEG_SLOPE 0.2f

// ---------------------------------------------------------------------------
// Utility kernels
// ---------------------------------------------------------------------------
__global__ void fill_f32(float* __restrict__ p, float val, int n) {
    int i = blockIdx.x * blockDim.x + threadIdx.x;
    if (i < n) p[i] = val;
}

// out[i] = b0[i % HD] + b1[i % HD]   (broadcast summed relation biases)
__global__ void init_bias(float* __restrict__ out, const float* __restrict__ b0,
                          const float* __restrict__ b1, int HD, int total) {
    int i = blockIdx.x * blockDim.x + threadIdx.x;
    if (i < total) {
        int t = i % HD;
        out[i] = b0[t] + b1[t];
    }
}

__global__ void elu_inplace(float* __restrict__ p, int n) {
    int i = blockIdx.x * blockDim.x + threadIdx.x;
    if (i < n) {
        float v = p[i];
        p[i] = (v > 0.0f) ? v : (__expf(v) - 1.0f);
    }
}

// ---------------------------------------------------------------------------
// WMMA f32 GEMM with LDS-staged W:  OUT[M, TN*16] = X[M, KK] @ W[KK, TN*16]
//  - one block = 8 waves; W (KK x TN*16) staged once into LDS
//  - each wave owns one 16-row stripe and keeps TN column-tile accumulators
//    in registers (TN compile-time so acc[] stays in VGPRs)
//  - V_WMMA_F32_16X16X4_F32 along K; one A-fragment load feeds TN WMMAs
// ---------------------------------------------------------------------------
template <int TN, int KK>
__global__ void gemm_f32_wmma_lds(const float* __restrict__ X, const float* __restrict__ W,
                                  float* __restrict__ OUT, int M) {
    constexpr int N = TN * 16;
    __shared__ float Wl[KK * N];

    // cooperative stage of W into LDS (fully coalesced b32/b128 loads)
    for (int i = threadIdx.x; i < KK * N; i += blockDim.x) Wl[i] = W[i];
    __syncthreads();

    int wave = threadIdx.x >> 5;
    int lane = threadIdx.x & 31;
    int rowTile = blockIdx.x * 8 + wave;
    if (rowTile * 16 >= M) return;            // wave-uniform: EXEC stays all-ones

    int half = lane >> 4;                     // 0: K,K+1   1: K+2,K+3
    int l    = lane & 15;

    v8f acc[TN] = {};
    const float* xrow = X + (size_t)(rowTile * 16 + l) * KK;

    for (int k = 0; k < KK; k += 4) {
        int ka = k + half * 2;
        v2f a;
        a.x = xrow[ka];
        a.y = xrow[ka + 1];
#pragma unroll
        for (int c = 0; c < TN; ++c) {
            v2f b;
            b.x = Wl[ka * N + c * 16 + l];
            b.y = Wl[(ka + 1) * N + c * 16 + l];
            acc[c] = __builtin_amdgcn_wmma_f32_16x16x4_f32(
                /*neg_a=*/false, a, /*neg_b=*/false, b,
                /*c_mod=*/(short)0, acc[c], /*reuse_a=*/false, /*reuse_b=*/false);
        }
    }

    // C/D layout: VGPR i -> row M=i (lanes 0-15) / M=i+8 (lanes 16-31), N=lane%16
    int rbase = rowTile * 16 + half * 8;
#pragma unroll
    for (int c = 0; c < TN; ++c)
#pragma unroll
        for (int i = 0; i < 8; ++i)
            OUT[(size_t)(rbase + i) * N + c * 16 + l] = acc[c][i];
}

// ---------------------------------------------------------------------------
// Attention scores: el[n,h] = sum_d feat[n,h,d]*al[h,d]  (er likewise)
// ---------------------------------------------------------------------------
__global__ void attn_scores(const float* __restrict__ feat,
                            const float* __restrict__ al, const float* __restrict__ ar,
                            float* __restrict__ el, float* __restrict__ er,
                            int Nn, int H, int D) {
    int i = blockIdx.x * blockDim.x + threadIdx.x;
    if (i >= Nn * H) return;
    int n = i / H, h = i % H;
    const float* f = feat + (size_t)n * H * D + h * D;
    const float* a = al + h * D;
    const float* r = ar + h * D;
    float sl = 0.0f, sr = 0.0f;
    for (int d = 0; d < D; ++d) {
        float v = f[d];
        sl += v * a[d];
        sr += v * r[d];
    }
    el[i] = sl;
    er[i] = sr;
}

// ---------------------------------------------------------------------------
// Edge passes (segment softmax + message aggregation)
// ---------------------------------------------------------------------------
__device__ __forceinline__ float leaky(float v) {
    return (v > 0.0f) ? v : NEG_SLOPE * v;
}

__device__ __forceinline__ void atomicMaxF(float* addr, float val) {
    unsigned* ua = (unsigned*)addr;
    unsigned old = *ua;
    while (__uint_as_float(old) < val) {
        unsigned assumed = old;
        old = atomicCAS(ua, assumed, __float_as_uint(val));
        if (old == assumed) break;
    }
}

// pass A: m[dst,h] = max over edges of leaky(el[src,h]+er[dst,h])
__global__ void edge_max_k(const float* __restrict__ el, const float* __restrict__ er,
                           const int* __restrict__ src, const int* __restrict__ dst,
                           float* __restrict__ m, int E, int H) {
    int i = blockIdx.x * blockDim.x + threadIdx.x;
    if (i >= E * H) return;
    int e = i / H, h = i % H;
    int sN = src[e], dN = dst[e];
    float v = leaky(el[sN * H + h] + er[dN * H + h]);
    atomicMaxF(&m[dN * H + h], v);
}

// pass B: s[dst,h] += exp(e - m[dst,h])
__global__ void edge_expsum_k(const float* __restrict__ el, const float* __restrict__ er,
                              const float* __restrict__ m,
                              const int* __restrict__ src, const int* __restrict__ dst,
                              float* __restrict__ s, int E, int H) {
    int i = blockIdx.x * blockDim.x + threadIdx.x;
    if (i >= E * H) return;
    int e = i / H, h = i % H;
    int sN = src[e], dN = dst[e];
    float v = leaky(el[sN * H + h] + er[dN * H + h]);
    atomicAdd(&s[dN * H + h], __expf(v - m[dN * H + h]));
}

// pass C: out[dst, h, d] += alpha * feat[src, h, d]
// one thread per (edge, 4-channel group): float4 gather (global_load_b128) +
// 4x global_atomic_add_f32; per-edge scalars fetched once per 4 channels.
// Channel groups never straddle a head boundary (D % 4 == 0).
__global__ void edge_msg_k(const float* __restrict__ feat,
                           const float* __restrict__ el, const float* __restrict__ er,
                           const float* __restrict__ m, const float* __restrict__ s,
                           const int* __restrict__ src, const int* __restrict__ dst,
                           float* __restrict__ out, int E, int H, int D) {
    int Q = (H * D) >> 2;                     // 4-channel groups per edge
    unsigned i = blockIdx.x * (unsigned)blockDim.x + threadIdx.x;
    if (i >= (unsigned)E * (unsigned)Q) return;
    int e = i / Q, q = i % Q;
    int t = q << 2, h = t / D;
    int sN = src[e], dN = dst[e];
    float v  = leaky(el[sN * H + h] + er[dN * H + h]);
    float ex = __expf(v - m[dN * H + h]);
    float alpha = ex / s[dN * H + h];
    const float4* f4 = (const float4*)(feat + (size_t)sN * H * D);
    float4 f = f4[q];
    float* ob = out + (size_t)dN * H * D + t;
    atomicAdd(ob + 0, alpha * f.x);
    atomicAdd(ob + 1, alpha * f.y);
    atomicAdd(ob + 2, alpha * f.z);
    atomicAdd(ob + 3, alpha * f.w);
}

// ---------------------------------------------------------------------------
// Orchestration
// ---------------------------------------------------------------------------
static inline unsigned nblk(long long n) { return (unsigned)((n + 255) / 256); }

extern "C" void kernel_launch(void* const* d_in, const int* in_sizes, int n_in,
                              void* d_out, int out_size, void* d_ws, size_t ws_size,
                              hipStream_t stream) {
    const int Nn = 50000, E = 800000;
    const float* x   = (const float*)d_in[0];    // [N,128]
    const float* W1  = (const float*)d_in[1];    // [2,128,128]
    const float* al1 = (const float*)d_in[2];    // [2,4,32]
    const float* ar1 = (const float*)d_in[3];
    const float* b1  = (const float*)d_in[4];    // [2,128]
    const float* W2  = (const float*)d_in[5];    // [2,128,64]
    const float* al2 = (const float*)d_in[6];    // [2,1,64]
    const float* ar2 = (const float*)d_in[7];
    const float* b2  = (const float*)d_in[8];    // [2,64]
    const int* src   = (const int*)d_in[9];      // [2,E]
    const int* dst   = (const int*)d_in[10];     // [2,E]
    float* out = (float*)d_out;                  // [N,64]

    // workspace layout (floats)
    float* ws   = (float*)d_ws;
    float* feat = ws;                       // 50000*128 (reused for layer2 [N,64])
    float* h1   = feat + (size_t)Nn * 128;  // 50000*128
    float* el   = h1 + (size_t)Nn * 128;    // 50000*4
    float* er   = el + Nn * 4;
    float* mbuf = er + Nn * 4;
    float* sbuf = mbuf + Nn * 4;
    const float NEG_INF = -3.402823466e38f;

    const int rowTiles = Nn / 16;                        // 3125
    const unsigned gemmBlocks = (unsigned)((rowTiles + 7) / 8);

    // ---------------- Layer 1: H=4, D=32, Fin=128, Fout=128 ----------------
    init_bias<<<nblk((long long)Nn * 128), 256, 0, stream>>>(h1, b1, b1 + 128, 128, Nn * 128);
    for (int r = 0; r < 2; ++r) {
        const int* sr = src + (size_t)r * E;
        const int* dr = dst + (size_t)r * E;
        gemm_f32_wmma_lds<8, 128><<<gemmBlocks, 256, 0, stream>>>(
            x, W1 + (size_t)r * 128 * 128, feat, Nn);
        attn_scores<<<nblk(Nn * 4), 256, 0, stream>>>(
            feat, al1 + r * 128, ar1 + r * 128, el, er, Nn, 4, 32);
        fill_f32<<<nblk(Nn * 4), 256, 0, stream>>>(mbuf, NEG_INF, Nn * 4);
        fill_f32<<<nblk(Nn * 4), 256, 0, stream>>>(sbuf, 0.0f, Nn * 4);
        edge_max_k<<<nblk((long long)E * 4), 256, 0, stream>>>(el, er, sr, dr, mbuf, E, 4);
        edge_expsum_k<<<nblk((long long)E * 4), 256, 0, stream>>>(el, er, mbuf, sr, dr, sbuf, E, 4);
        edge_msg_k<<<nblk((long long)E * 32), 256, 0, stream>>>(
            feat, el, er, mbuf, sbuf, sr, dr, h1, E, 4, 32);
    }
    elu_inplace<<<nblk((long long)Nn * 128), 256, 0, stream>>>(h1, Nn * 128);

    // ---------------- Layer 2: H=1, D=64, Fin=128, Fout=64 -----------------
    init_bias<<<nblk((long long)Nn * 64), 256, 0, stream>>>(out, b2, b2 + 64, 64, Nn * 64);
    for (int r = 0; r < 2; ++r) {
        const int* sr = src + (size_t)r * E;
        const int* dr = dst + (size_t)r * E;
        gemm_f32_wmma_lds<4, 128><<<gemmBlocks, 256, 0, stream>>>(
            h1, W2 + (size_t)r * 128 * 64, feat, Nn);
        attn_scores<<<nblk(Nn), 256, 0, stream>>>(
            feat, al2 + r * 64, ar2 + r * 64, el, er, Nn, 1, 64);
        fill_f32<<<nblk(Nn), 256, 0, stream>>>(mbuf, NEG_INF, Nn);
        fill_f32<<<nblk(Nn), 256, 0, stream>>>(sbuf, 0.0f, Nn);
        edge_max_k<<<nblk((long long)E), 256, 0, stream>>>(el, er, sr, dr, mbuf, E, 1);
        edge_expsum_k<<<nblk((long long)E), 256, 0, stream>>>(el, er, mbuf, sr, dr, sbuf, E, 1);
        edge_msg_k<<<nblk((long long)E * 16), 256, 0, stream>>>(
            feat, el, er, mbuf, sbuf, sr, dr, out, E, 1, 64);
    }
    // mean over head dim: H=1 -> identity, out already holds [N,64]
}